// FasterRCNN_27204322853181
// MI455X (gfx1250) — compile-verified
//
#include <hip/hip_runtime.h>
#include <stdint.h>

// ---------------- problem constants (match reference) ----------------
constexpr int   N      = 8;
constexpr int   A      = 15;          // len(SIZES)*len(RATIOS)
constexpr int   H      = 256;
constexpr int   W      = 256;
constexpr int   HW     = H * W;       // 65536
constexpr int   M      = HW * A;      // 983040 anchors per image
constexpr int   K      = 2000;        // pre_nms_top_n
constexpr int   KPAD   = 2048;        // sort padding (power of two)
constexpr int   KCH    = 2048;        // padded mask rows per image
constexpr int   NBINS  = 2048;        // score histogram bins over sigmoid in [0,1)
constexpr int   CBW    = 64;          // 64 * 32 = 2048 suppression bits per row
constexpr float IMGSZ  = 1024.0f;
constexpr float NMS_T  = 0.7f;
constexpr float NEGV   = -1000000000.0f;
constexpr float BBOX_CLIP = 4.135166556742356f;  // log(1000/16)

// ---------------- CDNA5 feature selection ----------------
#if __has_builtin(__builtin_amdgcn_tensor_load_to_lds) && __has_builtin(__builtin_amdgcn_s_wait_tensorcnt)
#define USE_TDM 1
#endif
#if __has_builtin(__builtin_amdgcn_wmma_f32_16x16x4_f32)
#define USE_WMMA 1
#endif

typedef unsigned int v4u __attribute__((ext_vector_type(4)));
typedef int          v8i __attribute__((ext_vector_type(8)));
typedef int          v4i __attribute__((ext_vector_type(4)));
typedef float        v2f __attribute__((ext_vector_type(2)));
typedef float        v8f __attribute__((ext_vector_type(8)));

// one b128 async copy global->LDS (ASYNCcnt-tracked)
__device__ __forceinline__ void async_copy_b128(unsigned lds_byte_addr, const void* gaddr) {
  unsigned long long ga = (unsigned long long)(uintptr_t)gaddr;
  asm volatile("global_load_async_to_lds_b128 %0, %1, off"
               :: "v"(lds_byte_addr), "v"(ga) : "memory");
}

__device__ __forceinline__ void async_wait0() {
#if __has_builtin(__builtin_amdgcn_s_wait_asynccnt)
  __builtin_amdgcn_s_wait_asynccnt(0);
#else
  asm volatile("s_wait_asynccnt 0" ::: "memory");
#endif
}

// TDM: DMA a 64-row x 64-dword tile (row stride CBW dwords) into LDS. TENSORcnt-tracked.
__device__ __forceinline__ void chunk_load(unsigned lds_byte_addr, const unsigned* gtile) {
#ifdef USE_TDM
  unsigned long long ga = (unsigned long long)(uintptr_t)gtile;
  v4u g0;
  g0.x = 1u;                                             // count=1, user mode
  g0.y = lds_byte_addr;                                  // D#.lds_addr
  g0.z = (unsigned)(ga & 0xFFFFFFFFull);                 // global_addr[31:0]
  g0.w = (unsigned)((ga >> 32) & 0x1FFFFFFull) | (2u << 30); // addr[56:32] | type=2
  v8i g1;
  g1[0] = (int)(2u << 16);                  // data_size=2 (4B), wg_mask=0, no barrier/iter/pad
  g1[1] = (int)((unsigned)CBW << 16);       // tensor_dim0[15:0] @ bits[63:48] = 64 elems
  g1[2] = (int)(64u << 16);                 // tensor_dim1[15:0] @ bits[95:80] = 64 rows
  g1[3] = (int)((unsigned)CBW << 16);       // tile_dim0 @ bits[127:112] = 64
  g1[4] = 64;                               // tile_dim1 @ bits[143:128] = 64, tile_dim2=0
  g1[5] = CBW;                              // tensor_dim0_stride[31:0] = 64 elems
  g1[6] = 0;                                // stride hi / dim1_stride
  g1[7] = 0;
  v4i z4 = {0, 0, 0, 0};
#if defined(__clang_major__) && (__clang_major__ >= 23)
  v8i z8 = {0, 0, 0, 0, 0, 0, 0, 0};
  __builtin_amdgcn_tensor_load_to_lds(g0, g1, z4, z4, z8, 0);   // 6-arg form
#else
  __builtin_amdgcn_tensor_load_to_lds(g0, g1, z4, z4, 0);       // 5-arg form (ROCm 7.2)
#endif
#else
  // fallback: 64 rows * 16 b128 each via async-LDS loads
  int lane = threadIdx.x & 31;
  for (int t = lane; t < 64 * 16; t += 32) {
    int byteoff = t * 16;
    async_copy_b128(lds_byte_addr + (unsigned)byteoff,
                    (const char*)gtile + byteoff);
  }
#endif
}

template <int NW>
__device__ __forceinline__ void chunk_wait() {
#ifdef USE_TDM
  __builtin_amdgcn_s_wait_tensorcnt((short)NW);
#else
  asm volatile("s_wait_asynccnt %0" :: "i"(NW) : "memory");
#endif
}

// ---------------- shared decode helper ----------------
__device__ __forceinline__ void decode_one(const float* __restrict__ reg,
                                           const float* __restrict__ anchors,
                                           int n, int i, float4& box, bool& valid) {
  int a    = i % A;
  int cell = i / A;                       // h*W + w
  const float* ap = anchors + (size_t)i * 4;
  float ax0 = ap[0], ay0 = ap[1], ax1 = ap[2], ay1 = ap[3];
  float w  = ax1 - ax0, h = ay1 - ay0;
  float cx = ax0 + 0.5f * w, cy = ay0 + 0.5f * h;
  size_t rb = ((size_t)n * A * 4 + (size_t)a * 4) * HW + (size_t)cell;
  float dx = reg[rb];
  float dy = reg[rb + (size_t)HW];
  float dw = fminf(reg[rb + 2 * (size_t)HW], BBOX_CLIP);
  float dh = fminf(reg[rb + 3 * (size_t)HW], BBOX_CLIP);
  float pcx = dx * w + cx, pcy = dy * h + cy;
  float pw = __expf(dw) * w, ph = __expf(dh) * h;
  float x0 = pcx - 0.5f * pw, y0 = pcy - 0.5f * ph;
  float x1 = pcx + 0.5f * pw, y1 = pcy + 0.5f * ph;
  x0 = fminf(fmaxf(x0, 0.0f), IMGSZ);  x1 = fminf(fmaxf(x1, 0.0f), IMGSZ);
  y0 = fminf(fmaxf(y0, 0.0f), IMGSZ);  y1 = fminf(fmaxf(y1, 0.0f), IMGSZ);
  box = make_float4(x0, y0, x1, y1);
  valid = (x1 - x0 >= 1.0f) && (y1 - y0 >= 1.0f);
}

// ---------------- kernels ----------------
__global__ void init_kernel(int* hist, int* meta, float* topS, int* topI, int* keep) {
  int gid = blockIdx.x * blockDim.x + threadIdx.x;
  if (gid < N * NBINS) hist[gid] = 0;
  if (gid < N * 8)     meta[gid] = 0;
  if (gid < N * KPAD)  { topS[gid] = NEGV; topI[gid] = 0; }
  if (gid < N * K)     keep[gid] = 0;
}

__global__ void decode_score_kernel(const float* __restrict__ obj,
                                    const float* __restrict__ reg,
                                    const float* __restrict__ anchors,
                                    float* __restrict__ probs) {
  size_t gid = (size_t)blockIdx.x * blockDim.x + threadIdx.x;
  if (gid >= (size_t)N * M) return;
  int n = (int)(gid / M), i = (int)(gid % M);
  int a = i % A, cell = i / A;
  float s = obj[((size_t)n * A + a) * HW + cell];
  float4 box; bool valid;
  decode_one(reg, anchors, n, i, box, valid);
  float p = 1.0f / (1.0f + __expf(-s));
  probs[gid] = valid ? p : NEGV;
}

constexpr int HBLK = 60;  // histogram blocks per image; M/HBLK = 16384 exactly
__global__ void hist_kernel(const float* __restrict__ probs, int* __restrict__ hist) {
  __shared__ int lh[NBINS];
  for (int b = threadIdx.x; b < NBINS; b += blockDim.x) lh[b] = 0;
  __syncthreads();
  int n = blockIdx.x / HBLK, c = blockIdx.x % HBLK;
  const int chunk = M / HBLK;
  const float* p = probs + (size_t)n * M + (size_t)c * chunk;
  for (int i = threadIdx.x; i < chunk; i += blockDim.x) {
    float v = p[i];
    if (v > -1.0e8f) {
      int b = (int)(v * (float)NBINS);
      b = b < 0 ? 0 : (b > NBINS - 1 ? NBINS - 1 : b);
      atomicAdd(&lh[b], 1);
    }
  }
  __syncthreads();
  for (int b = threadIdx.x; b < NBINS; b += blockDim.x)
    if (lh[b]) atomicAdd(&hist[n * NBINS + b], lh[b]);
}

__global__ void thresh_kernel(const int* __restrict__ hist, int* __restrict__ meta) {
  if (threadIdx.x != 0) return;
  int n = blockIdx.x;
  const int* h = hist + n * NBINS;
  int acc = 0, T = -1, Cab = 0;
  for (int b = NBINS - 1; b >= 0; --b) {
    int c = h[b];
    if (acc + c >= K) { T = b; Cab = acc; break; }
    acc += c;
  }
  meta[n * 8 + 0] = T;
  meta[n * 8 + 1] = (T < 0) ? 0 : Cab;
}

__global__ void compact_kernel(const float* __restrict__ probs, int* __restrict__ meta,
                               float* __restrict__ topS, int* __restrict__ topI) {
  size_t gid = (size_t)blockIdx.x * blockDim.x + threadIdx.x;
  if (gid >= (size_t)N * M) return;
  int n = (int)(gid / M), i = (int)(gid % M);
  float v = probs[gid];
  if (v <= -1.0e8f) return;
  int b = (int)(v * (float)NBINS);
  b = b < 0 ? 0 : (b > NBINS - 1 ? NBINS - 1 : b);
  int T = meta[n * 8 + 0], Cab = meta[n * 8 + 1];
  int slot = -1;
  if (b > T) {
    slot = atomicAdd(&meta[n * 8 + 2], 1);           // always < K by construction
  } else if (b == T) {
    slot = Cab + atomicAdd(&meta[n * 8 + 3], 1);
  }
  if (slot >= 0 && slot < K) {
    topS[n * KPAD + slot] = v;
    topI[n * KPAD + slot] = i;
  }
}

__global__ void __launch_bounds__(1024)
sort_kernel(const float* __restrict__ topS, const int* __restrict__ topI,
            float* __restrict__ srtS, int* __restrict__ srtI) {
  __shared__ float s[KPAD];
  __shared__ int   si[KPAD];
  int n = blockIdx.x;
  for (int i = threadIdx.x; i < KPAD; i += blockDim.x) {
    s[i] = topS[n * KPAD + i]; si[i] = topI[n * KPAD + i];
  }
  __syncthreads();
  for (unsigned k = 2; k <= (unsigned)KPAD; k <<= 1) {
    for (unsigned j = k >> 1; j > 0; j >>= 1) {
      for (unsigned i = threadIdx.x; i < (unsigned)KPAD; i += blockDim.x) {
        unsigned ixj = i ^ j;
        if (ixj > i) {
          float a = s[i], b = s[ixj];
          int ia = si[i], ib = si[ixj];
          bool aAfter = (a < b) || (a == b && ia > ib);   // descending, idx tiebreak
          bool descSeg = ((i & k) == 0);
          if (descSeg ? aAfter : !aAfter) {
            s[i] = b; s[ixj] = a; si[i] = ib; si[ixj] = ia;
          }
        }
      }
      __syncthreads();
    }
  }
  for (int i = threadIdx.x; i < K; i += blockDim.x) {
    srtS[n * KPAD + i] = s[i]; srtI[n * KPAD + i] = si[i];
  }
}

__global__ void decode_topk_kernel(const float* __restrict__ reg,
                                   const float* __restrict__ anchors,
                                   const int* __restrict__ srtI,
                                   float* __restrict__ boxes) {
  int gid = blockIdx.x * blockDim.x + threadIdx.x;
  if (gid >= N * K) return;
  int n = gid / K, k = gid % K;
  int idx = srtI[n * KPAD + k];
  float4 box; bool valid;
  decode_one(reg, anchors, n, idx, box, valid);
  float* o = boxes + ((size_t)n * K + k) * 4;
  o[0] = box.x; o[1] = box.y; o[2] = box.z; o[3] = box.w;
}

// ---- NMS pairwise mask: wave-tiled, WMMA computes (area_i + area_j) per tile.
// iou > T  <=>  (1+T)*inter > T*(area_i + area_j)   (degenerate union: both false)
constexpr int RTILES = K / 16;                 // 125 row tiles
constexpr int JWORDS = (K + 31) / 32;          // 63 col words
constexpr int WAVES_PER_BLOCK = 8;             // 256 threads
constexpr int TASKS  = RTILES * JWORDS;        // 7875 waves per image
constexpr int BPI    = (TASKS + WAVES_PER_BLOCK - 1) / WAVES_PER_BLOCK;  // 985

__global__ void __launch_bounds__(256)
nms_mask_kernel(const float* __restrict__ boxes, unsigned* __restrict__ mask) {
  __shared__ float sb[K * 4];                  // 32000 B: all K boxes of this image
  int n = blockIdx.x / BPI;
  const float* g = boxes + (size_t)n * K * 4;
  // stage boxes into LDS via async b128 copies (ASYNCcnt path)
  for (int r = threadIdx.x; r < K; r += blockDim.x) {
    unsigned lds = (unsigned)(uintptr_t)(&sb[r * 4]);
    async_copy_b128(lds, g + (size_t)r * 4);
  }
  async_wait0();
  __syncthreads();

  int wid  = threadIdx.x >> 5;
  int lane = threadIdx.x & 31;
  int task = (blockIdx.x % BPI) * WAVES_PER_BLOCK + wid;
  if (task >= TASKS) return;                   // uniform per wave: EXEC stays full
  int rt = task / JWORDS;                      // row tile
  int j  = task % JWORDS;                      // 32-col word
  int r0 = rt * 16;

  // A operand (16x4 f32): lanes 0-15 hold K=0,1 for row M=lane; lanes 16-31 = K=2,3 (zero)
  int ridx = r0 + (lane & 15);
  float rX0 = sb[ridx * 4], rY0 = sb[ridx * 4 + 1];
  float rX1 = sb[ridx * 4 + 2], rY1 = sb[ridx * 4 + 3];
  float areaR = (rX1 - rX0) * (rY1 - rY0);
  v2f Aop;
  Aop.x = (lane < 16) ? areaR : 0.0f;          // A[m][0] = area_m
  Aop.y = (lane < 16) ? 1.0f  : 0.0f;          // A[m][1] = 1

  unsigned ball[2][8];
  #pragma unroll
  for (int h = 0; h < 2; ++h) {
    int cb   = j * 32 + h * 16;
    int cidx = cb + (lane & 15);
    int cld  = cidx < K ? cidx : K - 1;
    float cX0 = sb[cld * 4], cY0 = sb[cld * 4 + 1];
    float cX1 = sb[cld * 4 + 2], cY1 = sb[cld * 4 + 3];
    float areaC = (cX1 - cX0) * (cY1 - cY0);
    // B operand (4x16 f32): lanes 0-15 hold K=0,1 rows; lanes 16-31 hold K=2,3 (zero)
    v2f Bop;
    Bop.x = (lane < 16) ? 1.0f  : 0.0f;        // B[0][n] = 1
    Bop.y = (lane < 16) ? areaC : 0.0f;        // B[1][n] = area_n
    v8f d;
#ifdef USE_WMMA
    v8f cz = {0.0f, 0.0f, 0.0f, 0.0f, 0.0f, 0.0f, 0.0f, 0.0f};
    d = __builtin_amdgcn_wmma_f32_16x16x4_f32(false, Aop, false, Bop,
                                              (short)0, cz, false, false);
#endif
    #pragma unroll
    for (int v = 0; v < 8; ++v) {
      int row = r0 + v + ((lane >= 16) ? 8 : 0);
      float x0 = sb[row * 4], y0 = sb[row * 4 + 1];
      float x1 = sb[row * 4 + 2], y1 = sb[row * 4 + 3];
      float ix0 = fmaxf(x0, cX0), iy0 = fmaxf(y0, cY0);
      float ix1 = fminf(x1, cX1), iy1 = fminf(y1, cY1);
      float iw = fmaxf(ix1 - ix0, 0.0f), ih = fmaxf(iy1 - iy0, 0.0f);
      float inter = iw * ih;
#ifdef USE_WMMA
      float ssum = d[v];                       // area_row + area_col from WMMA
#else
      float ssum = (x1 - x0) * (y1 - y0) + areaC;
#endif
      bool bit = (cidx < K) && (cidx > row) &&
                 (inter * (1.0f + NMS_T) > NMS_T * ssum);
      ball[h][v] = (unsigned)__ballot((int)bit);   // bits 0-15: row r0+v; 16-31: row r0+8+v
    }
  }
  // lanes 0-15 assemble + store the 32-bit word for row r0+lane
  int r = lane & 15;
  unsigned m0 = 0, m1 = 0;
  #pragma unroll
  for (int v = 0; v < 8; ++v) {
    m0 |= (r == v)     ? (ball[0][v] & 0xFFFFu) : 0u;
    m0 |= (r == v + 8) ? (ball[0][v] >> 16)     : 0u;
    m1 |= (r == v)     ? (ball[1][v] & 0xFFFFu) : 0u;
    m1 |= (r == v + 8) ? (ball[1][v] >> 16)     : 0u;
  }
  unsigned word = m0 | (m1 << 16);
  int row = r0 + r;
  if (lane < 16 && row < K)
    mask[((size_t)n * KCH + row) * CBW + j] = word;
}

// single wave32 per image; suppression bitmap lives in 2 VGPRs/lane;
// mask tiles streamed to LDS by TDM (TENSORcnt) double-buffered.
__global__ void __launch_bounds__(32)
nms_scan_kernel(const unsigned* __restrict__ mask, int* __restrict__ keep) {
  constexpr int RC = 64;                         // rows per chunk
  __shared__ unsigned ch[2][RC * CBW];           // 2 x 16 KB
  int n = blockIdx.x;
  int lane = threadIdx.x;
  const unsigned* mbase = mask + (size_t)n * KCH * CBW;
  unsigned r0 = 0, r1 = 0;                       // removed words: lane, lane+32
  const int nch = (K + RC - 1) / RC;             // 32 (mask padded to KCH rows)
  chunk_load((unsigned)(uintptr_t)(&ch[0][0]), mbase);
  for (int c = 0; c < nch; ++c) {
    if (c + 1 < nch) {
      chunk_load((unsigned)(uintptr_t)(&ch[(c + 1) & 1][0]),
                 mbase + (size_t)(c + 1) * RC * CBW);
      chunk_wait<1>();                           // chunk c complete (in-order)
    } else {
      chunk_wait<0>();
    }
    const unsigned* rowp = &ch[c & 1][0];
    int i0 = c * RC;
    int imax = (K - i0 < RC) ? (K - i0) : RC;
    for (int r = 0; r < imax; ++r) {
      int i = i0 + r;
      int w = i >> 5;
      unsigned cand = (w < 32) ? r0 : r1;
      unsigned rw = __shfl(cand, w & 31, 32);    // broadcast removed word i>>5
      if (((rw >> (i & 31)) & 1u) == 0u) {       // row i survives -> suppress later
        r0 |= rowp[r * CBW + lane];
        r1 |= rowp[r * CBW + 32 + lane];
      }
    }
  }
  // each lane owns bits for rows [lane*32, lane*32+32) and [(lane+32)*32, ...)
  for (int b = 0; b < 32; ++b) {
    int i = lane * 32 + b;
    if (i < K) keep[n * K + i] = ((r0 >> b) & 1u) ? 0 : 1;
    int i2 = (lane + 32) * 32 + b;
    if (i2 < K) keep[n * K + i2] = ((r1 >> b) & 1u) ? 0 : 1;
  }
}

__global__ void output_kernel(const float* __restrict__ boxes,
                              const float* __restrict__ srtS,
                              const int* __restrict__ keep,
                              float* __restrict__ out) {
  int gid = blockIdx.x * blockDim.x + threadIdx.x;
  if (gid >= N * K) return;
  int n = gid / K, k = gid % K;
  float kf = keep[n * K + k] ? 1.0f : 0.0f;
  const float* b = boxes + ((size_t)n * K + k) * 4;
  float* o = out + (size_t)gid * 5;
  o[0] = b[0] * kf; o[1] = b[1] * kf; o[2] = b[2] * kf; o[3] = b[3] * kf;
  o[4] = srtS[n * KPAD + k] * kf;
}

// ---------------- launcher ----------------
extern "C" void kernel_launch(void* const* d_in, const int* in_sizes, int n_in,
                              void* d_out, int out_size, void* d_ws, size_t ws_size,
                              hipStream_t stream) {
  const float* obj     = (const float*)d_in[0];
  const float* reg     = (const float*)d_in[1];
  const float* anchors = (const float*)d_in[2];
  float* out = (float*)d_out;
  char* ws = (char*)d_ws;

  size_t off = 0;
  auto take = [&](size_t bytes) {
    size_t cur = off;
    off += (bytes + 255) & ~(size_t)255;
    return cur;
  };
  float*    probs = (float*)   (ws + take(sizeof(float)   * (size_t)N * M));
  int*      hist  = (int*)     (ws + take(sizeof(int)     * (size_t)N * NBINS));
  int*      meta  = (int*)     (ws + take(sizeof(int)     * (size_t)N * 8));
  float*    topS  = (float*)   (ws + take(sizeof(float)   * (size_t)N * KPAD));
  int*      topI  = (int*)     (ws + take(sizeof(int)     * (size_t)N * KPAD));
  float*    srtS  = (float*)   (ws + take(sizeof(float)   * (size_t)N * KPAD));
  int*      srtI  = (int*)     (ws + take(sizeof(int)     * (size_t)N * KPAD));
  float*    boxes = (float*)   (ws + take(sizeof(float)   * (size_t)N * K * 4));
  unsigned* mask  = (unsigned*)(ws + take(sizeof(unsigned)* (size_t)N * KCH * CBW));
  int*      keep  = (int*)     (ws + take(sizeof(int)     * (size_t)N * K));
  (void)in_sizes; (void)n_in; (void)out_size; (void)ws_size;

  const int total = N * M;                       // 7,864,320
  init_kernel        <<<(N * KPAD + 255) / 256, 256, 0, stream>>>(hist, meta, topS, topI, keep);
  decode_score_kernel<<<(total + 255) / 256, 256, 0, stream>>>(obj, reg, anchors, probs);
  hist_kernel        <<<N * HBLK, 256, 0, stream>>>(probs, hist);
  thresh_kernel      <<<N, 32, 0, stream>>>(hist, meta);
  compact_kernel     <<<(total + 255) / 256, 256, 0, stream>>>(probs, meta, topS, topI);
  sort_kernel        <<<N, 1024, 0, stream>>>(topS, topI, srtS, srtI);
  decode_topk_kernel <<<(N * K + 255) / 256, 256, 0, stream>>>(reg, anchors, srtI, boxes);
  nms_mask_kernel    <<<N * BPI, 256, 0, stream>>>(boxes, mask);
  nms_scan_kernel    <<<N, 32, 0, stream>>>(mask, keep);
  output_kernel      <<<(N * K + 255) / 256, 256, 0, stream>>>(boxes, srtS, keep, out);
}